// Network_7490422964931
// MI455X (gfx1250) — compile-verified
//
#include <hip/hip_runtime.h>
#include <math.h>

// MI455X / gfx1250, wave32. SG-visibility kernel.
// Roofline: 36 MB traffic (~1.6us @ 23.3 TB/s) vs ~3.4 GFLOP with 4
// transcendentals per (n,l,s) triple -> instruction-throughput bound.
// The 6->16 MLP layer (96 FMAs/lane naively) runs on the matrix pipe:
//   D[hidden=16][elem=16] = W1^T (16x6, K zero-padded to 8) x X^T (6x16)
// 2 element-tiles per wave x 2 K-chunks = 4x V_WMMA_F32_16X16X4_F32.
// Marshalling exploits that points are block-uniform (n = blockIdx>>2), so
// only the 3 input_dir components need a xor-16 exchange (ds_swizzle,
// compile-time SWAPX16 pattern, no index setup). Per-lobe frames (normalize,
// acos, sin/cos of r_phi) are computed once per block by warp 0 into LDS
// (32 lobes x 12 floats), amortizing ~50 VALU+3 trans per lane.

typedef float v2f __attribute__((ext_vector_type(2)));
typedef float v8f __attribute__((ext_vector_type(8)));

#define TINYV 1e-6f
#define LLOB 128
#define SSAM 8
#define HHID 16
#define NELEM (8192 * 128 * 8)     // 8,388,608 — divisible by 256

// ds_swizzle group-of-32 patterns: offset = (xor<<10)|(or<<5)|and
#define SWZ(x, pat) __int_as_float(__builtin_amdgcn_ds_swizzle(__float_as_int(x), (pat)))
#define PAT_XOR16 0x401F
#define PAT_XOR1  0x041F
#define PAT_XOR2  0x081F
#define PAT_XOR4  0x101F

__global__ __launch_bounds__(256) void sg_vis_wmma_kernel(
    const float* __restrict__ points,   // (N,3)
    const float* __restrict__ normals,  // (N,3)
    const float* __restrict__ rootR,    // (3,3) row-major
    const float* __restrict__ lobes,    // (L,3)
    const float* __restrict__ lambdas,  // (L,1)
    const float* __restrict__ rrand,    // (N,L,S)
    const float* __restrict__ W1,       // (6,16) row-major
    const float* __restrict__ b1,       // (16)
    const float* __restrict__ W2,       // (16,1)
    const float* __restrict__ b2,       // (1)
    float* __restrict__ out)            // (N,L)
{
    __shared__ float sF[32 * 12];       // 32 lobes x 11 floats (stride 12 for b128 loads)

    const int tid  = threadIdx.x;
    const int e    = blockIdx.x * 256 + tid;   // element id, s fastest
    const int lane = tid & 31;

    // ---- warp 0: build the 32 lobe frames this block needs ----
    if (tid < 32) {
        const int l = ((blockIdx.x * 32) + tid) & (LLOB - 1);
        float lx = lobes[3*l+0], ly = lobes[3*l+1], lz = lobes[3*l+2];
        float invn = 1.f / (sqrtf(lx*lx + ly*ly + lz*lz) + TINYV);
        lx *= invn; ly *= invn; lz *= invn;
        // U = norm(cross(z, l)) = norm(-ly, lx, 0)  (U.z == 0)
        float ux = -ly, uy = lx;
        float invu = 1.f / (sqrtf(ux*ux + uy*uy) + TINYV);
        ux *= invu; uy *= invu;
        // V = norm(cross(l, U)) = norm(-lz*uy, lz*ux, lx*uy - ly*ux)
        float vx = -lz*uy, vy = lz*ux, vz = lx*uy - ly*ux;
        float invv = 1.f / (sqrtf(vx*vx + vy*vy + vz*vz) + TINYV);
        vx *= invv; vy *= invv; vz *= invv;
        float sharp = lambdas[l];
        float rphi  = fminf(acosf(1.f - 1.f/sharp), 1.04719755119659775f); // pi/3
        float* F = &sF[tid * 12];
        F[0] = lx; F[1] = ly; F[2] = lz;
        F[3] = ux; F[4] = uy;
        F[5] = vx; F[6] = vy; F[7] = vz;
        F[8] = __sinf(rphi); F[9] = __cosf(rphi); F[10] = sharp; F[11] = 0.f;
    }
    __syncthreads();

    // ---- per-lane: load frame, build sample direction ----
    const float* F = &sF[(tid >> 3) * 12];
    const float lnx = F[0], lny = F[1], lnz = F[2];
    const float ux  = F[3], uy  = F[4];
    const float vx  = F[5], vy  = F[6], vz = F[7];
    const float sp  = F[8], cp  = F[9], sharp = F[10];

    const float theta = (rrand[e] * (1.f/SSAM) + (float)(tid & 7) * (1.f/SSAM))
                        * 6.28318530717958648f;
    const float st = __sinf(theta), ct = __cosf(theta);
    const float ca = ct * sp, cb = st * sp;
    const float dx = ux*ca + vx*cb + lnx*cp;
    const float dy = uy*ca + vy*cb + lny*cp;
    const float dz =          vz*cb + lnz*cp;          // U.z == 0

    // points/normals are uniform per block (1024 elems per n) -> scalar loads
    const int n = blockIdx.x >> 2;
    const float px = points[3*n+0],  py = points[3*n+1],  pz = points[3*n+2];
    const float nx = normals[3*n+0], ny = normals[3*n+1], nz = normals[3*n+2];
    const bool cosmask = (nx*dx + ny*dy + nz*dz) > TINYV;

    // input_dir = -(dir^T * R)
    const float id0 = -(dx*rootR[0] + dy*rootR[3] + dz*rootR[6]);
    const float id1 = -(dx*rootR[1] + dy*rootR[4] + dz*rootR[7]);
    const float id2 = -(dx*rootR[2] + dy*rootR[5] + dz*rootR[8]);

    const float w = __expf(sharp * (dx*lnx + dy*lny + dz*lnz - 1.f));

    // ---- MLP layer 1 via WMMA: D = W1^T (16x[4|4]) x X^T ([4|4]x16) ----
    const int  m     = lane & 15;
    const bool up    = lane >= 16;
    const int  hbase = up ? 8 : 0;

    // A = W1^T: lane m holds hidden row M=m; lower half K=0,1 / upper K=2,3.
    v2f A0, A1;
    A0.x = up ? W1[2*HHID + m] : W1[0*HHID + m];
    A0.y = up ? W1[3*HHID + m] : W1[1*HHID + m];
    A1.x = up ? 0.f : W1[4*HHID + m];   // K=4 (upper K=6 pad = 0)
    A1.y = up ? 0.f : W1[5*HHID + m];   // K=5 (upper K=7 pad = 0)

    // X columns: K0..2 = point (block-uniform!), K3..5 = input_dir.
    // Only id0/id1/id2 need the cross-half exchange.
    const float x3 = id0, x4 = id1, x5 = id2;
    const float y3 = SWZ(x3, PAT_XOR16);
    const float y4 = SWZ(x4, PAT_XOR16);
    const float y5 = SWZ(x5, PAT_XOR16);

    v2f B0t0, B1t0, B0t1, B1t1;
    // tile0 = elements 0..15: upper lanes carry K2,K3 of elem (lane-16)
    B0t0.x = up ? pz  : px;   B0t0.y = up ? y3  : py;
    B1t0.x = up ? 0.f : x4;   B1t0.y = up ? 0.f : x5;
    // tile1 = elements 16..31: lower lanes carry K0,K1 (uniform point)
    B0t1.x = up ? pz  : px;   B0t1.y = up ? x3  : py;
    B1t1.x = up ? 0.f : y4;   B1t1.y = up ? 0.f : y5;

    // C seeded with bias: lane holds hidden rows hbase..hbase+7
    v8f c0, c1;
    #pragma unroll
    for (int r = 0; r < 8; ++r) c0[r] = b1[hbase + r];
    c1 = c0;

    c0 = __builtin_amdgcn_wmma_f32_16x16x4_f32(false, A0, false, B0t0, (short)0, c0, false, false);
    c0 = __builtin_amdgcn_wmma_f32_16x16x4_f32(false, A1, false, B1t0, (short)0, c0, false, false);
    c1 = __builtin_amdgcn_wmma_f32_16x16x4_f32(false, A0, false, B0t1, (short)0, c1, false, false);
    c1 = __builtin_amdgcn_wmma_f32_16x16x4_f32(false, A1, false, B1t1, (short)0, c1, false, false);

    // ---- layer 2: ReLU + 16-tap dot. Lane computes partials for both tiles;
    //      sends the one its xor-16 partner needs (one swizzle total). ----
    float w2v[8];
    #pragma unroll
    for (int r = 0; r < 8; ++r) w2v[r] = W2[hbase + r];
    float pa = 0.f, pb = 0.f;
    #pragma unroll
    for (int r = 0; r < 8; ++r) {
        const float t0 = c0[r], t1 = c1[r];
        pa += (t0 > 0.f ? t0 : 0.f) * w2v[r];
        pb += (t1 > 0.f ? t1 : 0.f) * w2v[r];
    }
    const float q  = up ? pa : pb;          // partial the partner's tile needs
    const float qq = SWZ(q, PAT_XOR16);
    const float logit = (up ? pb : pa) + qq + b2[0];
    const float pred  = 1.f / (1.f + __expf(-logit));
    const float vis   = cosmask ? pred : 0.f;

    // ---- reduce over s (8 contiguous lanes) ----
    float num = vis * w, den = w;
    num += SWZ(num, PAT_XOR1);  den += SWZ(den, PAT_XOR1);
    num += SWZ(num, PAT_XOR2);  den += SWZ(den, PAT_XOR2);
    num += SWZ(num, PAT_XOR4);  den += SWZ(den, PAT_XOR4);

    if ((lane & 7) == 0) out[e >> 3] = num / (den + TINYV);
}

extern "C" void kernel_launch(void* const* d_in, const int* in_sizes, int n_in,
                              void* d_out, int out_size, void* d_ws, size_t ws_size,
                              hipStream_t stream) {
    const float* points  = (const float*)d_in[0];
    const float* normals = (const float*)d_in[1];
    const float* rootR   = (const float*)d_in[2];
    const float* lobes   = (const float*)d_in[3];
    const float* lambdas = (const float*)d_in[4];
    const float* rrand   = (const float*)d_in[5];
    const float* W1      = (const float*)d_in[6];
    const float* b1      = (const float*)d_in[7];
    const float* W2      = (const float*)d_in[8];
    const float* b2      = (const float*)d_in[9];
    float* out = (float*)d_out;

    dim3 grid(NELEM / 256), block(256);
    sg_vis_wmma_kernel<<<grid, block, 0, stream>>>(
        points, normals, rootR, lobes, lambdas, rrand, W1, b1, W2, b2, out);
}